// ModernBertCanonLayer_50843822850517
// MI455X (gfx1250) — compile-verified
//
#include <hip/hip_runtime.h>
#include <stdint.h>

// Depthwise varlen conv (K=5) + bias + residual for packed sequences.
// T=65536, C=768, fp32. Memory-bound (~403 MB moved, ~17 us floor @ 23.3 TB/s).
// CDNA5 path: async global->LDS staging (global_load_async_to_lds_b128,
// s_wait_asynccnt), ds_load_b128 tap reuse, non-temporal b128 stores.

typedef float v4f __attribute__((ext_vector_type(4)));

namespace {
constexpr int T_TOT = 65536;
constexpr int C     = 768;
constexpr int KT    = 5;
constexpr int RAD   = 2;
constexpr int NCU   = 17;           // cu_seqlens entries (0 .. T)
constexpr int TILE  = 16;           // rows per block
constexpr int HROWS = TILE + 2 * RAD;   // 20 staged rows
constexpr int NTHR  = C / 4;        // 192 threads: one float4 channel group each
}

__global__ __launch_bounds__(NTHR)
void canon_dwconv_kernel(const float* __restrict__ x,
                         const float* __restrict__ w,      // [C, K]
                         const float* __restrict__ bias,   // [C]
                         const int*   __restrict__ cu,     // [17]
                         float* __restrict__ out)          // [T, C]
{
    __shared__ float xs[HROWS * C];     // 61440 B staged tile (rows t0-2 .. t0+17)
    __shared__ int   se[2 * TILE];      // per-row [seq_start, seq_end)

    const int tid = threadIdx.x;
    const int c   = tid * 4;
    const int t0  = (int)blockIdx.x * TILE;

    // ---- stage x tile into LDS with async copies (ASYNCcnt-tracked) ----
    #pragma unroll
    for (int r = 0; r < HROWS; ++r) {
        int row = t0 - RAD + r;
        row = row < 0 ? 0 : (row >= T_TOT ? T_TOT - 1 : row);  // clamped; masked at compute
        uint64_t gaddr = (uint64_t)(uintptr_t)(x + (size_t)row * C + c);
        unsigned laddr = (unsigned)(uintptr_t)(&xs[r * C + c]); // low 32b of flat = LDS offset
        asm volatile("global_load_async_to_lds_b128 %0, %1, off"
                     :: "v"(laddr), "v"(gaddr) : "memory");
    }

    // ---- per-row sequence bounds: s = max cu[i] <= t, e = min cu[i] > t ----
    if (tid < TILE) {
        const int t = t0 + tid;
        int s = 0, e = T_TOT;
        #pragma unroll
        for (int i = 1; i < NCU - 1; ++i) {
            const int b = cu[i];            // cu sorted ascending
            if (b <= t) s = b;
            else        e = (b < e) ? b : e;
        }
        se[2 * tid + 0] = s;
        se[2 * tid + 1] = e;
    }

    // ---- per-thread tap weights + bias held in registers for all TILE rows ----
    float wk[KT][4];
    #pragma unroll
    for (int k = 0; k < KT; ++k) {
        #pragma unroll
        for (int j = 0; j < 4; ++j) {
            wk[k][j] = w[(c + j) * KT + k];
        }
    }
    float b4[4];
    #pragma unroll
    for (int j = 0; j < 4; ++j) b4[j] = bias[c + j];

    asm volatile("s_wait_asynccnt 0" ::: "memory");
    __syncthreads();

    // ---- compute: 5-tap MAC from LDS, residual add, NT b128 store ----
    for (int r = 0; r < TILE; ++r) {
        const int t = t0 + r;
        const int s = se[2 * r + 0];
        const int e = se[2 * r + 1];

        float acc0 = b4[0], acc1 = b4[1], acc2 = b4[2], acc3 = b4[3];
        const float* xrow = &xs[(r + RAD) * C + c];

        if ((t - RAD >= s) && (t + RAD < e)) {
            // interior fast path (wave-uniform branch): all 5 taps valid
            #pragma unroll
            for (int k = 0; k < KT; ++k) {
                const v4f xv = *(const v4f*)(xrow + (k - RAD) * C);
                acc0 = fmaf(xv.x, wk[k][0], acc0);
                acc1 = fmaf(xv.y, wk[k][1], acc1);
                acc2 = fmaf(xv.z, wk[k][2], acc2);
                acc3 = fmaf(xv.w, wk[k][3], acc3);
            }
        } else {
            // boundary path: per-tap validity (covers global edges too,
            // since cu[0]=0 and cu[last]=T)
            #pragma unroll
            for (int k = 0; k < KT; ++k) {
                const int tt = t + (k - RAD);
                if (tt >= s && tt < e) {
                    const v4f xv = *(const v4f*)(xrow + (k - RAD) * C);
                    acc0 = fmaf(xv.x, wk[k][0], acc0);
                    acc1 = fmaf(xv.y, wk[k][1], acc1);
                    acc2 = fmaf(xv.z, wk[k][2], acc2);
                    acc3 = fmaf(xv.w, wk[k][3], acc3);
                }
            }
        }

        const v4f xc = *(const v4f*)xrow;   // residual (center row)
        v4f o4;
        o4.x = xc.x + acc0;
        o4.y = xc.y + acc1;
        o4.z = xc.z + acc2;
        o4.w = xc.w + acc3;
        __builtin_nontemporal_store(o4, (v4f*)(out + (size_t)t * C + c));
    }
}

extern "C" void kernel_launch(void* const* d_in, const int* in_sizes, int n_in,
                              void* d_out, int out_size, void* d_ws, size_t ws_size,
                              hipStream_t stream) {
    (void)in_sizes; (void)n_in; (void)out_size; (void)d_ws; (void)ws_size;
    const float* x  = (const float*)d_in[0];   // [T, C]
    const float* w  = (const float*)d_in[1];   // [C, K]
    const float* b  = (const float*)d_in[2];   // [C]
    const int*  cu  = (const int*) d_in[3];    // [17]
    float* out = (float*)d_out;                // [T, C]

    dim3 grid(T_TOT / TILE);   // 4096 blocks
    dim3 block(NTHR);          // 192 threads = 6 wave32s
    canon_dwconv_kernel<<<grid, block, 0, stream>>>(x, w, b, cu, out);
}